// GNN_45896020525803
// MI455X (gfx1250) — compile-verified
//
#include <hip/hip_runtime.h>
#include <math.h>

#define NUM_USER 50000
#define NUM_ITEM 50000
#define NN (NUM_USER + NUM_ITEM)
#define NE 1200000
#define DF 128
#define DL 64

typedef __attribute__((ext_vector_type(2))) float v2f;
typedef __attribute__((ext_vector_type(8))) float v8f;

__device__ __forceinline__ float lrelu(float x) { return x >= 0.f ? x : 0.01f * x; }

__device__ __forceinline__ float wave_sum(float v) {
#pragma unroll
    for (int o = 16; o > 0; o >>= 1) v += __shfl_xor(v, o, 32);
    return v;
}

// monotonic int mapping trick: valid with init = -INF
__device__ __forceinline__ void atomicMaxF(float* addr, float val) {
    if (val >= 0.f) atomicMax((int*)addr, __float_as_int(val));
    else            atomicMin((unsigned int*)addr, __float_as_uint(val));
}

__global__ void fill_kernel(float* __restrict__ p, float v, int n) {
    int i = blockIdx.x * blockDim.x + threadIdx.x;
    if (i < n) p[i] = v;
}

__global__ void copy_kernel(const float* __restrict__ s, float* __restrict__ d, int n) {
    int i = blockIdx.x * blockDim.x + threadIdx.x;
    if (i < n) d[i] = s[i];
}

// C[M,64] = post(act(A[M,K] @ W (+bias) (+addend_pre)))
// WT=1: W stored [64,K] (torch Linear weight -> W^T); WT=0: W stored [K,64]
// ACT: 0=none 1=tanh 2=leaky ; ADDM: 0=none 1=pre-act 2=post-act
// One wave computes a 16x64 output tile via V_WMMA_F32_16X16X4_F32, K fully unrolled.
template <int K, int WT, int ACT, int ADDM>
__global__ void gemm64_wmma(const float* __restrict__ A, int lda, int M,
                            const float* __restrict__ W,
                            const float* __restrict__ bias,
                            const float* __restrict__ addend, int add_ld,
                            float* __restrict__ C, int ldc, int col_off)
{
    int wave = (blockIdx.x * blockDim.x + threadIdx.x) >> 5;
    if (wave >= (M >> 4)) return;            // whole-wave uniform guard: EXEC stays full
    int lane = threadIdx.x & 31;
    int hf  = lane >> 4;                     // half-wave selects K pair
    int l16 = lane & 15;
    int row0 = wave << 4;

    v8f acc[4] = {};                         // 4 column tiles of 16 -> 64 outputs

    // hoisted per-lane base pointers; all loop offsets become immediates
    const float* arow = A + (size_t)(row0 + l16) * lda + (hf << 1);
    const float* wb0, *wb1, *wb2, *wb3;
    if (WT) {
        wb0 = W + (size_t)(0 * 16 + l16) * K + (hf << 1);
        wb1 = W + (size_t)(1 * 16 + l16) * K + (hf << 1);
        wb2 = W + (size_t)(2 * 16 + l16) * K + (hf << 1);
        wb3 = W + (size_t)(3 * 16 + l16) * K + (hf << 1);
    } else {
        wb0 = W + (size_t)(hf << 1) * 64 + 0 * 16 + l16;
        wb1 = W + (size_t)(hf << 1) * 64 + 1 * 16 + l16;
        wb2 = W + (size_t)(hf << 1) * 64 + 2 * 16 + l16;
        wb3 = W + (size_t)(hf << 1) * 64 + 3 * 16 + l16;
    }

#pragma unroll
    for (int kk = 0; kk < K; kk += 4) {
        v2f a;                               // A 16x4 frag: lane=row, VGPR0/1 = K pair
        a.x = arow[kk];
        a.y = arow[kk + 1];
        v2f b0, b1, b2, b3;                  // B 4x16 frags: row striped across lanes
        if (WT) {
            b0.x = wb0[kk]; b0.y = wb0[kk + 1];
            b1.x = wb1[kk]; b1.y = wb1[kk + 1];
            b2.x = wb2[kk]; b2.y = wb2[kk + 1];
            b3.x = wb3[kk]; b3.y = wb3[kk + 1];
        } else {
            b0.x = wb0[kk * 64]; b0.y = wb0[(kk + 1) * 64];
            b1.x = wb1[kk * 64]; b1.y = wb1[(kk + 1) * 64];
            b2.x = wb2[kk * 64]; b2.y = wb2[(kk + 1) * 64];
            b3.x = wb3[kk * 64]; b3.y = wb3[(kk + 1) * 64];
        }
        acc[0] = __builtin_amdgcn_wmma_f32_16x16x4_f32(false, a, false, b0, (short)0, acc[0], false, false);
        acc[1] = __builtin_amdgcn_wmma_f32_16x16x4_f32(false, a, false, b1, (short)0, acc[1], false, false);
        acc[2] = __builtin_amdgcn_wmma_f32_16x16x4_f32(false, a, false, b2, (short)0, acc[2], false, false);
        acc[3] = __builtin_amdgcn_wmma_f32_16x16x4_f32(false, a, false, b3, (short)0, acc[3], false, false);
    }

#pragma unroll
    for (int n = 0; n < 4; ++n) {
        int col = (n << 4) + l16;
        float bv = bias ? bias[col] : 0.f;
#pragma unroll
        for (int r = 0; r < 8; ++r) {
            int row = row0 + r + (hf << 3);  // C layout: VGPR r -> M=r (lanes<16) / r+8
            float v = acc[n][r] + bv;
            if (ADDM == 1) v += addend[(size_t)row * add_ld + col];
            if (ACT == 1)      v = tanhf(v);
            else if (ACT == 2) v = lrelu(v);
            if (ADDM == 2) v += addend[(size_t)row * add_ld + col];
            C[(size_t)row * ldc + col_off + col] = v;
        }
    }
}

// out = l2norm(in + bias), optionally leaky afterwards; one wave per node (64 dims)
__global__ void norm_kernel(const float* __restrict__ in, const float* __restrict__ bias,
                            float* __restrict__ outp, int n, int do_leaky)
{
    int w = (blockIdx.x * blockDim.x + threadIdx.x) >> 5;
    int lane = threadIdx.x & 31;
    if (w >= n) return;
    const v2f* in2 = (const v2f*)in;
    v2f* out2 = (v2f*)outp;
    size_t idx = (size_t)w * 32 + lane;
    v2f v = in2[idx];
    if (bias) { v.x += bias[2 * lane]; v.y += bias[2 * lane + 1]; }
    float ss = wave_sum(v.x * v.x + v.y * v.y);
    float inv = 1.f / fmaxf(sqrtf(ss), 1e-12f);
    v.x *= inv; v.y *= inv;
    if (do_leaky) { v.x = lrelu(v.x); v.y = lrelu(v.y); }
    out2[idx] = v;
}

__global__ void deg_kernel(const int* __restrict__ src, float* __restrict__ deg, int E) {
    int e = blockIdx.x * blockDim.x + threadIdx.x;
    if (e < E) atomicAdd(&deg[src[e]], 1.f);
}

// per-edge (one wave): inner = sum(x_i * leaky(x_j)); t = inner*sigmoid(deg^-.5*inner)
__global__ void edge_inner_kernel(const int* __restrict__ src, const int* __restrict__ tgt,
                                  const float* __restrict__ xw, const float* __restrict__ deg,
                                  float* __restrict__ t, float* __restrict__ m, int E)
{
    int e = (blockIdx.x * blockDim.x + threadIdx.x) >> 5;
    int lane = threadIdx.x & 31;
    if (e >= E) return;
    int s = src[e], g = tgt[e];
    const v2f* xw2 = (const v2f*)xw;
    v2f xj = xw2[(size_t)s * 32 + lane];
    v2f xi = xw2[(size_t)g * 32 + lane];
    float p = xi.x * lrelu(xj.x) + xi.y * lrelu(xj.y);
    p = wave_sum(p);
    if (lane == 0) {
        float dis  = rsqrtf(deg[s]);
        float gate = 1.f / (1.f + expf(-dis * p));
        float tv   = p * gate;
        t[e] = tv;
        atomicMaxF(&m[g], tv);
    }
}

__global__ void edge_exp_kernel(const int* __restrict__ tgt, const float* __restrict__ t,
                                const float* __restrict__ m, float* __restrict__ em,
                                float* __restrict__ ssum, int E)
{
    int e = blockIdx.x * blockDim.x + threadIdx.x;
    if (e >= E) return;
    int g = tgt[e];
    float ex = expf(t[e] - m[g]);
    em[e] = ex;
    atomicAdd(&ssum[g], ex);
}

__global__ void edge_agg_kernel(const int* __restrict__ src, const int* __restrict__ tgt,
                                const float* __restrict__ xw, const float* __restrict__ em,
                                const float* __restrict__ ssum, float* __restrict__ agg, int E)
{
    int e = (blockIdx.x * blockDim.x + threadIdx.x) >> 5;
    int lane = threadIdx.x & 31;
    if (e >= E) return;
    int s = src[e], g = tgt[e];
    float att = em[e] / (ssum[g] + 1e-16f);
    const v2f* xw2 = (const v2f*)xw;
    v2f xj = xw2[(size_t)s * 32 + lane];
    size_t gb = (size_t)g * 64 + 2 * lane;
    atomicAdd(&agg[gb],     xj.x * att);
    atomicAdd(&agg[gb + 1], xj.y * att);
}

template <int K, int WT, int ACT, int ADDM>
static void gemm(hipStream_t st, const float* A, int lda, int M,
                 const float* W, const float* bias,
                 const float* addend, int add_ld,
                 float* C, int ldc, int col_off)
{
    int tiles = M / 16;
    int blocks = (tiles + 3) / 4;   // 4 waves / block
    gemm64_wmma<K, WT, ACT, ADDM><<<blocks, 128, 0, st>>>(
        A, lda, M, W, bias, addend, add_ld, C, ldc, col_off);
}

static void run_gat(hipStream_t st, const float* x, int x_ld,
                    const float* gw, const float* gb,
                    const int* src, const int* tgt,
                    float* xw, float* agg, float* h_out,
                    float* t, float* em, float* deg, float* mmax, float* ssum)
{
    // xw = x @ gw   (gw stored [d_in, d_out] -> no transpose)
    gemm<DL, 0, 0, 0>(st, x, x_ld, NN, gw, nullptr, nullptr, 0, xw, DL, 0);
    fill_kernel<<<(NN + 255) / 256, 256, 0, st>>>(deg, 0.f, NN);
    fill_kernel<<<(NN + 255) / 256, 256, 0, st>>>(mmax, -INFINITY, NN);
    fill_kernel<<<(NN + 255) / 256, 256, 0, st>>>(ssum, 0.f, NN);
    fill_kernel<<<(NN * DL + 255) / 256, 256, 0, st>>>(agg, 0.f, NN * DL);
    deg_kernel<<<(NE + 255) / 256, 256, 0, st>>>(src, deg, NE);
    edge_inner_kernel<<<(NE + 7) / 8, 256, 0, st>>>(src, tgt, xw, deg, t, mmax, NE);
    edge_exp_kernel<<<(NE + 255) / 256, 256, 0, st>>>(tgt, t, mmax, em, ssum, NE);
    edge_agg_kernel<<<(NE + 7) / 8, 256, 0, st>>>(src, tgt, xw, em, ssum, agg, NE);
    // h = leaky(l2norm(agg + b))
    norm_kernel<<<(NN + 7) / 8, 256, 0, st>>>(agg, gb, h_out, NN, 1);
}

extern "C" void kernel_launch(void* const* d_in, const int* in_sizes, int n_in,
                              void* d_out, int out_size, void* d_ws, size_t ws_size,
                              hipStream_t stream)
{
    const float* features   = (const float*)d_in[0];
    const float* preference = (const float*)d_in[1];
    const float* id_emb     = (const float*)d_in[2];
    const float* mlp_w      = (const float*)d_in[3];
    const float* mlp_b      = (const float*)d_in[4];
    const float* gat1_w     = (const float*)d_in[5];
    const float* gat1_b     = (const float*)d_in[6];
    const float* lin1_w     = (const float*)d_in[7];
    const float* lin1_b     = (const float*)d_in[8];
    const float* g1_w       = (const float*)d_in[9];
    const float* g1_b       = (const float*)d_in[10];
    const float* gat2_w     = (const float*)d_in[11];
    const float* gat2_b     = (const float*)d_in[12];
    const float* lin2_w     = (const float*)d_in[13];
    const float* lin2_b     = (const float*)d_in[14];
    const float* g2_w       = (const float*)d_in[15];
    const float* g2_b       = (const float*)d_in[16];
    const int*   ei         = (const int*)d_in[17];
    const int*   src        = ei;            // edge_index[0]
    const int*   tgt        = ei + NE;       // edge_index[1]
    float* out = (float*)d_out;              // [N, 128] = concat(x_1, x_2)

    float* ws = (float*)d_ws;
    size_t nb = (size_t)NN * DL;
    float* bufA = ws;                        // x
    float* bufB = bufA + nb;                 // xw / x_hat
    float* bufC = bufB + nb;                 // agg / h
    float* tbuf = bufC + nb;                 // per-edge t
    float* em   = tbuf + NE;                 // per-edge exp
    float* deg  = em + NE;
    float* mmax = deg + NN;
    float* ssum = mmax + NN;

    // x = l2norm(concat(preference, tanh(features @ mlp_w^T + mlp_b)))
    copy_kernel<<<(NUM_USER * DL + 255) / 256, 256, 0, stream>>>(preference, bufA, NUM_USER * DL);
    gemm<DF, 1, 1, 0>(stream, features, DF, NUM_ITEM, mlp_w, mlp_b,
                      nullptr, 0, bufA + (size_t)NUM_USER * DL, DL, 0);
    norm_kernel<<<(NN + 7) / 8, 256, 0, stream>>>(bufA, nullptr, bufA, NN, 0);

    // ---- layer 1 ----
    run_gat(stream, bufA, DL, gat1_w, gat1_b, src, tgt, bufB, bufC, bufC,
            tbuf, em, deg, mmax, ssum);                         // h -> bufC
    // x_hat = leaky(x @ lin1^T + b) + id_embedding  -> bufB
    gemm<DL, 1, 2, 2>(stream, bufA, DL, NN, lin1_w, lin1_b, id_emb, DL, bufB, DL, 0);
    // x_1 = leaky(h @ g1^T + b + x_hat) -> out[:, 0:64]
    gemm<DL, 1, 2, 1>(stream, bufC, DL, NN, g1_w, g1_b, bufB, DL, out, 2 * DL, 0);

    // ---- layer 2 (reads x_1 from d_out with stride 128) ----
    run_gat(stream, out, 2 * DL, gat2_w, gat2_b, src, tgt, bufB, bufC, bufC,
            tbuf, em, deg, mmax, ssum);                         // h2 -> bufC
    // x_hat2 = leaky(x_1 @ lin2^T + b) + id_embedding -> bufB
    gemm<DL, 1, 2, 2>(stream, out, 2 * DL, NN, lin2_w, lin2_b, id_emb, DL, bufB, DL, 0);
    // x_2 = leaky(h2 @ g2^T + b + x_hat2) -> out[:, 64:128]
    gemm<DL, 1, 2, 1>(stream, bufC, DL, NN, g2_w, g2_b, bufB, DL, out, 2 * DL, DL);
}